// PowerSpectrum_34102040330735
// MI455X (gfx1250) — compile-verified
//
#include <hip/hip_runtime.h>

typedef __attribute__((ext_vector_type(2))) float v2f;
typedef __attribute__((ext_vector_type(8))) float v8f;

#define Q 64
#define OUT_PER_SN (4 * Q * Q)   // 16384 floats per (s,n)

// LDS staging layout: 24 rows x 64 floats (zero padded to K multiples of 4)
//   l=0 rows  0..3  (1 valid row)
//   l=1 rows  4..7  (3 valid rows)
//   l=2 rows  8..15 (5 valid rows)
//   l=3 rows 16..23 (7 valid rows)
__global__ __launch_bounds__(256) void ps_wmma_kernel(
    const float* __restrict__ c0, const float* __restrict__ c1,
    const float* __restrict__ c2, const float* __restrict__ c3,
    float* __restrict__ out)
{
    __shared__ float lds[24 * Q];   // 6 KB

    const int sn  = blockIdx.x;     // flattened (s,n)
    const int tid = threadIdx.x;    // 0..255 (8 waves of 32)

    // ---- stage inputs into LDS once, zero-padding the K dimension ----
    const long sb = (long)sn;
    #pragma unroll
    for (int k = 0; k < 6; ++k) {
        int idx  = tid + k * 256;       // 0..1535
        int prow = idx >> 6;            // padded row 0..23
        int q    = idx & 63;
        float v = 0.0f;
        if (prow < 4) {
            if (prow < 1) v = __builtin_nontemporal_load(&c0[sb * Q + q]);          // M=1
        } else if (prow < 8) {
            int m = prow - 4;  if (m < 3) v = __builtin_nontemporal_load(&c1[(sb * 3 + m) * Q + q]); // M=3
        } else if (prow < 16) {
            int m = prow - 8;  if (m < 5) v = __builtin_nontemporal_load(&c2[(sb * 5 + m) * Q + q]); // M=5
        } else {
            int m = prow - 16; if (m < 7) v = __builtin_nontemporal_load(&c3[(sb * 7 + m) * Q + q]); // M=7
        }
        lds[idx] = v;
    }
    __syncthreads();

    const int wave = tid >> 5;          // 0..7
    const int lane = tid & 31;
    const int lrow = lane & 15;         // M/N index within half-wave
    const int khi  = (lane >> 4) << 1;  // K offset: 0 (lanes 0-15) or 2 (lanes 16-31)
    const int hi8  = (lane >> 4) << 3;  // C/D row offset for upper half-wave

    const float INV_SQRT3 = 0.5773502691896258f;
    float* outbase = out + (size_t)sn * OUT_PER_SN;

    // Compile-time unrolled over l so kb/nc/g fold to immediates and the
    // K-chunk loop fully unrolls (no exec-masked dynamic loop around WMMA).
    #pragma unroll
    for (int l = 0; l < 4; ++l) {
        const int   kb = (l == 0) ? 0 : (l == 1) ? 4 : (l == 2) ? 8 : 16;
        const int   nc = (l < 2) ? 1 : 2;   // K chunks of 4 (zero-padded)
        const float g  = (l == 0) ? 1.0f
                       : (l == 1) ? 0.5773502691896258f   // 3^-1/2
                       : (l == 2) ? 0.4472135954999579f   // 5^-1/2
                       :            0.3779644730092272f;  // 7^-1/2
        float* lbase = outbase + l * (Q * Q);

        // 16 tiles per l-gram, 8 waves -> 2 tiles per wave
        #pragma unroll
        for (int rep = 0; rep < 2; ++rep) {
            const int t  = wave + rep * 8;  // 0..15
            const int ti = t >> 2;          // tile row
            const int tj = t & 3;           // tile col

            v8f acc = {};
            #pragma unroll
            for (int kc = 0; kc < nc; ++kc) {
                const int k0 = kb + kc * 4 + khi;
                v2f a, b;
                // A: rows = q-index (ti*16 + lrow); K striped across VGPR pair
                a.x = lds[(k0    ) * Q + ti * 16 + lrow];
                a.y = lds[(k0 + 1) * Q + ti * 16 + lrow];
                // B: cols = q-index (tj*16 + lrow); same K striping
                b.x = lds[(k0    ) * Q + tj * 16 + lrow];
                b.y = lds[(k0 + 1) * Q + tj * 16 + lrow];
                // D = A(16x4) * B(4x16) + C   (fp32 WMMA)
                acc = __builtin_amdgcn_wmma_f32_16x16x4_f32(
                    /*neg_a=*/false, a, /*neg_b=*/false, b,
                    /*c_mod=*/(short)0, acc, /*reuse_a=*/false, /*reuse_b=*/false);
            }

            // scale + non-temporal store:
            // acc[r] = element (row = ti*16 + 8*(lane>=16) + r, col = tj*16 + lane%16)
            const int col   = tj * 16 + lrow;
            const int rbase = ti * 16 + hi8;
            float* ob = lbase + col;
            #pragma unroll
            for (int r = 0; r < 8; ++r) {
                const int row = rbase + r;
                const float nrm = (row == col) ? g * INV_SQRT3 : g;
                __builtin_nontemporal_store(acc[r] * nrm, &ob[row * Q]);
            }
        }
    }
}

extern "C" void kernel_launch(void* const* d_in, const int* in_sizes, int n_in,
                              void* d_out, int out_size, void* d_ws, size_t ws_size,
                              hipStream_t stream) {
    const float* c0 = (const float*)d_in[0];
    const float* c1 = (const float*)d_in[1];
    const float* c2 = (const float*)d_in[2];
    const float* c3 = (const float*)d_in[3];
    float* out = (float*)d_out;

    // c0 has shape (S, N, 1, 64) -> number of (s,n) pairs = size/64
    const int SN = in_sizes[0] / Q;

    ps_wmma_kernel<<<SN, 256, 0, stream>>>(c0, c1, c2, c3, out);
}